// TransMLPWithGCNStylization_84318797955717
// MI455X (gfx1250) — compile-verified
//
#include <hip/hip_runtime.h>
#include <hip/hip_bf16.h>

#define B_  16
#define P_  12
#define D_  256
#define T_  50
#define TP_ 64                  // padded T for WMMA N-dim
#define KP_ 52                  // padded T for WMMA K-dim
#define NTOT (B_*P_*D_*T_)      // 2,457,600
#define EPS_ 1e-5f

typedef float v2f __attribute__((ext_vector_type(2)));
typedef float v8f __attribute__((ext_vector_type(8)));

// ---------------------------------------------------------------------------
// CDNA5 fp32 WMMA tile: D(16x16) += A(16x4) * B(4x16), exact fp32.
// Lane layout per ISA 7.12.2:
//   A: m = lane&15, ks = 2*(lane>>4); a.x=A[m][ks], a.y=A[m][ks+1]
//   B: n = lane&15, same ks;          b.x=B[ks][n], b.y=B[ks+1][n]
//   C/D: VGPR v -> row (v + 8*(lane>>4)), col lane&15
// ---------------------------------------------------------------------------
__device__ inline v8f wmma_f32_k4(v2f a, v2f b, v8f c) {
#if defined(__gfx1250__) && __has_builtin(__builtin_amdgcn_wmma_f32_16x16x4_f32)
  return __builtin_amdgcn_wmma_f32_16x16x4_f32(false, a, false, b, (short)0, c,
                                               false, false);
#else
  // host-pass / fallback placeholder (never used on gfx1250 device pass)
  c[0] += a.x * b.x + a.y * b.y;
  return c;
#endif
}

// ===========================================================================
// Kernel 1: temporal-FC (WMMA, M=D N=T K=T) + LayerNorm(D).
//   mode 0:  out = base + LN_D(A @ W^T + bvec) * alpha + beta   (MLP temporal)
//   mode 1:  out = LN_D(base + A @ W^T + bvec) * alpha + beta   (stylize/xg)
// One block (8 waves) per (b,p); each wave register-blocks 2 M-tiles x 4
// N-tiles (8 accumulators).  W is staged in LDS pre-swizzled to a
// K-PAIR-INTERLEAVED B-fragment layout Bsm[k/2][n][2] (zero-padded in K and
// N) so each B fragment is ONE aligned 8-byte ds_load straight into the
// operand register pair — no repack movs, no exec-masked loads.
// A-fragment K-tail over-reads (u=50,51) are annihilated by the zero B-pad
// and all A operands live inside d_ws, so the <=2-element over-read is benign.
// ===========================================================================
__global__ __launch_bounds__(256) void tfc_ln_kernel(
    const float* __restrict__ base, const float* __restrict__ A,
    const float* __restrict__ W, const float* __restrict__ bvec,
    const float* __restrict__ alpha, const float* __restrict__ beta,
    float* __restrict__ out, int mode)
{
  __shared__ float Hs[D_ * T_];        // 50 KB
  __shared__ float Bsm[KP_ * TP_];     // 13 KB, pair-interleaved B fragments
  __shared__ float s_mean[T_], s_rstd[T_];

  const int bp  = blockIdx.x;
  const int tid = threadIdx.x;
  // Bsm[(u>>1)*(2*TP_) + col*2 + (u&1)] = W[col][u]  (0 outside 50x50)
  for (int i = tid; i < KP_ * TP_; i += 256) {
    const int u2  = i / (2 * TP_);
    const int rem = i % (2 * TP_);
    const int col = rem >> 1;
    const int u   = 2 * u2 + (rem & 1);
    Bsm[i] = (u < T_ && col < T_) ? W[col * T_ + u] : 0.f;
  }
  __syncthreads();

  const int wave = tid >> 5, lane = tid & 31;
  const int l16 = lane & 15, lh = lane >> 4;
  const int m0 = wave * 32;                       // rows [m0, m0+32)
  const float* Abase = A    + (size_t)bp * D_ * T_;
  const float* Bbase = base + (size_t)bp * D_ * T_;
  const float* Ar0 = Abase + (size_t)(m0 + l16) * T_;
  const float* Ar1 = Abase + (size_t)(m0 + 16 + l16) * T_;

  v8f acc[2][4] = {};
  for (int kk = 0; kk < KP_; kk += 4) {
    const int ka = kk + 2 * lh;                   // even
    v2f a0, a1;
    a0.x = Ar0[ka];  a0.y = Ar0[ka + 1];   // tail garbage * B-pad zero = 0
    a1.x = Ar1[ka];  a1.y = Ar1[ka + 1];
    v2f bb[4];
#pragma unroll
    for (int tn = 0; tn < 4; ++tn) {
      const int col = tn * 16 + l16;
      bb[tn] = *(const v2f*)&Bsm[ka * TP_ + 2 * col];   // 8B-aligned pair
    }
#pragma unroll
    for (int tn = 0; tn < 4; ++tn) {
      acc[0][tn] = wmma_f32_k4(a0, bb[tn], acc[0][tn]);
      acc[1][tn] = wmma_f32_k4(a1, bb[tn], acc[1][tn]);
    }
  }
#pragma unroll
  for (int mi = 0; mi < 2; ++mi) {
#pragma unroll
    for (int tn = 0; tn < 4; ++tn) {
      const int col = tn * 16 + l16;
      if (col < T_) {
#pragma unroll
        for (int v = 0; v < 8; ++v) {
          const int m = m0 + mi * 16 + v + 8 * lh;
          float h = acc[mi][tn][v] + bvec[col];
          if (mode == 1) h += Bbase[(size_t)m * T_ + col];
          Hs[m * T_ + col] = h;
        }
      }
    }
  }
  __syncthreads();

  if (tid < T_) {
    float s = 0.f, s2 = 0.f;
    for (int d = 0; d < D_; ++d) {
      float h = Hs[d * T_ + tid];
      s += h; s2 += h * h;
    }
    float mean = s * (1.f / D_);
    float var  = s2 * (1.f / D_) - mean * mean;
    s_mean[tid] = mean;
    s_rstd[tid] = rsqrtf(var + EPS_);
  }
  __syncthreads();

  for (int idx = tid; idx < D_ * T_; idx += 256) {
    const int d = idx / T_, t = idx % T_;
    float ln = (Hs[idx] - s_mean[t]) * s_rstd[t] * alpha[d] + beta[d];
    out[(size_t)bp * D_ * T_ + idx] = (mode == 0 ? Bbase[idx] : 0.f) + ln;
  }
}

// ===========================================================================
// Kernel 2: spatial FC (WMMA fp32 D x D GEMM) + LN(D) + residual, per (b,p):
//   out = x + LN_D(Ws @ x + bs) * alpha + beta
// 8 waves; wave register-blocks 2 M-tiles x 4 N-tiles, K = D = 256.
// Tail N-tile handled by clamped index + mask multiply (no exec-mask loads).
// ===========================================================================
__global__ __launch_bounds__(256) void spatial_fc_ln_kernel(
    const float* __restrict__ xin, const float* __restrict__ Ws,
    const float* __restrict__ bs, const float* __restrict__ alpha,
    const float* __restrict__ beta, float* __restrict__ xout)
{
  __shared__ float Hs[D_ * T_];      // 50 KB
  __shared__ float s_mean[T_], s_rstd[T_];

  const int bp  = blockIdx.x;
  const int tid = threadIdx.x;
  const float* Xbp = xin + (size_t)bp * D_ * T_;

  const int wave = tid >> 5, lane = tid & 31;
  const int l16 = lane & 15, lh = lane >> 4;
  const int m0 = wave * 32;

  int   colc[4];
  float cm[4];
#pragma unroll
  for (int tn = 0; tn < 4; ++tn) {
    const int col = tn * 16 + l16;
    colc[tn] = (col < T_) ? col : (T_ - 1);
    cm[tn]   = (col < T_) ? 1.f : 0.f;
  }

  v8f acc[2][4] = {};
  for (int kk = 0; kk < D_; kk += 4) {
    const int ka = kk + 2 * lh;
    v2f a0, a1;
    a0.x = Ws[(size_t)(m0 + l16) * D_ + ka];
    a0.y = Ws[(size_t)(m0 + l16) * D_ + ka + 1];
    a1.x = Ws[(size_t)(m0 + 16 + l16) * D_ + ka];
    a1.y = Ws[(size_t)(m0 + 16 + l16) * D_ + ka + 1];
    v2f bb[4];
#pragma unroll
    for (int tn = 0; tn < 4; ++tn) {
      bb[tn].x = Xbp[(size_t)ka * T_ + colc[tn]]       * cm[tn];
      bb[tn].y = Xbp[(size_t)(ka + 1) * T_ + colc[tn]] * cm[tn];
    }
#pragma unroll
    for (int tn = 0; tn < 4; ++tn) {
      acc[0][tn] = wmma_f32_k4(a0, bb[tn], acc[0][tn]);
      acc[1][tn] = wmma_f32_k4(a1, bb[tn], acc[1][tn]);
    }
  }
#pragma unroll
  for (int mi = 0; mi < 2; ++mi) {
#pragma unroll
    for (int tn = 0; tn < 4; ++tn) {
      const int col = tn * 16 + l16;
      if (col < T_) {
#pragma unroll
        for (int v = 0; v < 8; ++v) {
          const int m = m0 + mi * 16 + v + 8 * lh;
          Hs[m * T_ + col] = acc[mi][tn][v] + bs[m];
        }
      }
    }
  }
  __syncthreads();

  if (tid < T_) {
    float s = 0.f, s2 = 0.f;
    for (int d = 0; d < D_; ++d) {
      float h = Hs[d * T_ + tid];
      s += h; s2 += h * h;
    }
    float mean = s * (1.f / D_);
    float var  = s2 * (1.f / D_) - mean * mean;
    s_mean[tid] = mean;
    s_rstd[tid] = rsqrtf(var + EPS_);
  }
  __syncthreads();

  for (int idx = tid; idx < D_ * T_; idx += 256) {
    const int d = idx / T_, t = idx % T_;
    float ln = (Hs[idx] - s_mean[t]) * s_rstd[t] * alpha[d] + beta[d];
    xout[(size_t)bp * D_ * T_ + idx] = Xbp[idx] + ln;
  }
}

// ===========================================================================
// Kernel 3: adjacency softmax over neighbours q.  One thread per (b,t,p).
// ===========================================================================
__global__ void adj_softmax_kernel(const float* __restrict__ dist,
                                   const int* __restrict__ mask,
                                   float* __restrict__ adj)
{
  const int i = blockIdx.x * blockDim.x + threadIdx.x;   // (b,t,p)
  if (i >= B_ * T_ * P_) return;
  const int b = i / (T_ * P_);
  const float* dr = dist + (size_t)i * P_;
  float lg[P_]; float mx = -1e30f;
#pragma unroll
  for (int q = 0; q < P_; ++q) {
    float v = (mask[b * P_ + q] == 0) ? -1e9f : -dr[q];
    lg[q] = v; mx = fmaxf(mx, v);
  }
  float s = 0.f;
#pragma unroll
  for (int q = 0; q < P_; ++q) { lg[q] = __expf(lg[q] - mx); s += lg[q]; }
  const float inv = 1.f / s;
  float* ar = adj + (size_t)i * P_;
#pragma unroll
  for (int q = 0; q < P_; ++q) ar[q] = lg[q] * inv;
}

// ===========================================================================
// Kernel 4: one GCN layer per (b,p):
//   AGG[d,t] = sum_q adj[b,t,p,q] * hin[b,q,d,t]   (staged in LDS)
//   hout     = relu(W @ AGG + bvec)  (WMMA, 2x4 register blocking),
//   optional final mask scale.  Tail tile via clamp + mask multiply.
// ===========================================================================
__global__ __launch_bounds__(256) void gcn_layer_kernel(
    const float* __restrict__ hin, const float* __restrict__ adj,
    const float* __restrict__ W, const float* __restrict__ bvec,
    const int* __restrict__ mask, float* __restrict__ hout, int apply_mask)
{
  __shared__ float AG[D_ * T_];      // 50 KB
  __shared__ float adjr[T_ * P_];    // 2.4 KB

  const int bp = blockIdx.x;
  const int b = bp / P_, p = bp % P_;
  const int tid = threadIdx.x;

  for (int i = tid; i < T_ * P_; i += 256) {
    const int t = i / P_, q = i % P_;
    adjr[i] = adj[(((size_t)b * T_ + t) * P_ + p) * P_ + q];
  }
  __syncthreads();

  for (int idx = tid; idx < D_ * T_; idx += 256) {
    const int d = idx / T_, t = idx % T_;
    float s = 0.f;
#pragma unroll
    for (int q = 0; q < P_; ++q)
      s += adjr[t * P_ + q] * hin[(((size_t)b * P_ + q) * D_ + d) * T_ + t];
    AG[idx] = s;
  }
  __syncthreads();

  const int wave = tid >> 5, lane = tid & 31;
  const int l16 = lane & 15, lh = lane >> 4;
  const int m0 = wave * 32;
  const float mval = apply_mask ? (float)mask[b * P_ + p] : 1.f;

  int   colc[4];
  float cm[4];
#pragma unroll
  for (int tn = 0; tn < 4; ++tn) {
    const int col = tn * 16 + l16;
    colc[tn] = (col < T_) ? col : (T_ - 1);
    cm[tn]   = (col < T_) ? 1.f : 0.f;
  }

  v8f acc[2][4] = {};
  for (int kk = 0; kk < D_; kk += 4) {
    const int ka = kk + 2 * lh;
    v2f a0, a1;
    a0.x = W[(size_t)(m0 + l16) * D_ + ka];
    a0.y = W[(size_t)(m0 + l16) * D_ + ka + 1];
    a1.x = W[(size_t)(m0 + 16 + l16) * D_ + ka];
    a1.y = W[(size_t)(m0 + 16 + l16) * D_ + ka + 1];
    v2f bb[4];
#pragma unroll
    for (int tn = 0; tn < 4; ++tn) {
      bb[tn].x = AG[ka * T_ + colc[tn]]       * cm[tn];
      bb[tn].y = AG[(ka + 1) * T_ + colc[tn]] * cm[tn];
    }
#pragma unroll
    for (int tn = 0; tn < 4; ++tn) {
      acc[0][tn] = wmma_f32_k4(a0, bb[tn], acc[0][tn]);
      acc[1][tn] = wmma_f32_k4(a1, bb[tn], acc[1][tn]);
    }
  }
#pragma unroll
  for (int mi = 0; mi < 2; ++mi) {
#pragma unroll
    for (int tn = 0; tn < 4; ++tn) {
      const int col = tn * 16 + l16;
      if (col < T_) {
#pragma unroll
        for (int v = 0; v < 8; ++v) {
          const int m = m0 + mi * 16 + v + 8 * lh;
          float y = fmaxf(acc[mi][tn][v] + bvec[m], 0.f);
          hout[(((size_t)b * P_ + p) * D_ + m) * T_ + col] = y * mval;
        }
      }
    }
  }
}

// ===========================================================================
// Kernel 5: person-attention pooling weights attn[b,p].
// ===========================================================================
__global__ __launch_bounds__(256) void attn_pool_kernel(
    const float* __restrict__ xg, const float* __restrict__ w_pool,
    const float* __restrict__ b_pool, const int* __restrict__ mask,
    float* __restrict__ attn)
{
  __shared__ float part[P_ * 16];
  __shared__ float sc[P_];
  const int b = blockIdx.x, tid = threadIdx.x;

  if (tid < P_ * 16) {
    const int p = tid >> 4, r = tid & 15;
    float s = 0.f;
    for (int d = r; d < D_; d += 16) {
      const float* row = xg + (((size_t)b * P_ + p) * D_ + d) * T_;
      float m = 0.f;
#pragma unroll
      for (int t = 0; t < T_; ++t) m += row[t];
      s += (m * (1.f / T_)) * w_pool[d];
    }
    part[tid] = s;
  }
  __syncthreads();
  if (tid < P_) {
    float s = b_pool[0];
#pragma unroll
    for (int r = 0; r < 16; ++r) s += part[tid * 16 + r];
    sc[tid] = (mask[b * P_ + tid] == 0) ? -1e9f : s;
  }
  __syncthreads();
  if (tid < P_) {
    float mx = -1e30f;
#pragma unroll
    for (int q = 0; q < P_; ++q) mx = fmaxf(mx, sc[q]);
    float sum = 0.f;
#pragma unroll
    for (int q = 0; q < P_; ++q) sum += __expf(sc[q] - mx);
    attn[b * P_ + tid] = __expf(sc[tid] - mx) / sum;
  }
}

// ===========================================================================
// Kernel 6: pooled global feature -> (scale, shift) via w_emb [2T,T].
// One block per b; thread d owns feature row d.
// ===========================================================================
__global__ __launch_bounds__(256) void scale_shift_kernel(
    const float* __restrict__ xg, const float* __restrict__ attn,
    const float* __restrict__ w_emb, const float* __restrict__ b_emb,
    float* __restrict__ scale, float* __restrict__ shift)
{
  __shared__ float aw[P_];
  __shared__ float Wemb[2 * T_ * T_];   // 20 KB
  const int b = blockIdx.x, d = threadIdx.x;
  for (int i = d; i < 2 * T_ * T_; i += 256) Wemb[i] = w_emb[i];
  if (d < P_) aw[d] = attn[b * P_ + d];
  __syncthreads();

  float xr[T_];
#pragma unroll
  for (int t = 0; t < T_; ++t) xr[t] = 0.f;
  for (int p = 0; p < P_; ++p) {
    const float* row = xg + (((size_t)b * P_ + p) * D_ + d) * T_;
    const float a = aw[p];
#pragma unroll
    for (int t = 0; t < T_; ++t) xr[t] += a * row[t];
  }
  for (int j = 0; j < T_; ++j) {
    float s1 = b_emb[j], s2 = b_emb[T_ + j];
#pragma unroll
    for (int t = 0; t < T_; ++t) {
      s1 += xr[t] * Wemb[j * T_ + t];
      s2 += xr[t] * Wemb[(T_ + j) * T_ + t];
    }
    scale[((size_t)b * D_ + d) * T_ + j] = s1;
    shift[((size_t)b * D_ + d) * T_ + j] = s2;
  }
}

// ===========================================================================
// Kernel 7: x1 = (x*(1+scale)+shift) @ w_out^T + b_out   (T-FC per row)
// ===========================================================================
__global__ __launch_bounds__(256) void modulate_fc_kernel(
    const float* __restrict__ x, const float* __restrict__ scale,
    const float* __restrict__ shift, const float* __restrict__ w_out,
    const float* __restrict__ b_out, float* __restrict__ out)
{
  __shared__ float Wo[T_ * T_];
  const int bp = blockIdx.x, b = bp / P_, d = threadIdx.x;
  for (int i = d; i < T_ * T_; i += 256) Wo[i] = w_out[i];
  __syncthreads();

  const float* xr = x     + ((size_t)bp * D_ + d) * T_;
  const float* sc = scale + ((size_t)b  * D_ + d) * T_;
  const float* sh = shift + ((size_t)b  * D_ + d) * T_;
  float m[T_];
#pragma unroll
  for (int t = 0; t < T_; ++t) m[t] = xr[t] * (1.f + sc[t]) + sh[t];
  for (int t = 0; t < T_; ++t) {
    float s = b_out[t];
#pragma unroll
    for (int u = 0; u < T_; ++u) s += m[u] * Wo[t * T_ + u];
    out[((size_t)bp * D_ + d) * T_ + t] = s;
  }
}

// ===========================================================================
// Kernel 8 (dominant GEMM): fused distance encoding + attention + context.
// One block per (b,p,t), 16 waves. A-matrix h1 (12x256, relu(s*w_d1+b_d1))
// is synthesized in registers — the 118 MB h1/d_enc tensors never exist.
// The A-synthesis VALU co-executes with the XDL-class WMMA pipe.
//   E(16x256) = h1 @ w_d2^T + b_d2   (one 16x16 tile/wave, K=256)
//   al[q] = E[q,:] . w_attn + b_attn ; aw = softmax_q ; dctx = aw @ E
// ===========================================================================
__global__ __launch_bounds__(512) void denc_kernel(
    const float* __restrict__ dist, const int* __restrict__ mask,
    const float* __restrict__ w_d1, const float* __restrict__ b_d1,
    const float* __restrict__ w_d2, const float* __restrict__ b_d2,
    const float* __restrict__ w_attn, const float* __restrict__ b_attn,
    float* __restrict__ dctx)
{
  __shared__ float E[16 * D_];          // 16 KB
  __shared__ float sv[16];
  __shared__ float wd1s[D_], bd1s[D_];  // 2 KB
  __shared__ float al[16], aw[P_];

  const int blk = blockIdx.x;           // b*P*T + p*T + t
  const int t  = blk % T_;
  const int bp = blk / T_;
  const int b = bp / P_, p = bp % P_;
  const int tid = threadIdx.x;

  if (tid < 16)
    sv[tid] = (tid < P_) ? dist[(((size_t)b * T_ + t) * P_ + p) * P_ + tid] : 0.f;
  for (int i = tid; i < D_; i += 512) { wd1s[i] = w_d1[i]; bd1s[i] = b_d1[i]; }
  __syncthreads();

  const int wave = tid >> 5, lane = tid & 31;
  const int l16 = lane & 15, lh = lane >> 4;
  const int e0 = wave * 16;             // this wave's 16 output features
  const float s_m = sv[l16];
  const float mvalid = (l16 < P_) ? 1.f : 0.f;

  v8f acc = {};
  for (int kk = 0; kk < D_; kk += 4) {
    const int ka = kk + 2 * lh;
    v2f av, bv;
    av.x = fmaxf(s_m * wd1s[ka]     + bd1s[ka],     0.f) * mvalid;
    av.y = fmaxf(s_m * wd1s[ka + 1] + bd1s[ka + 1], 0.f) * mvalid;
    bv.x = w_d2[(size_t)(e0 + l16) * D_ + ka];
    bv.y = w_d2[(size_t)(e0 + l16) * D_ + ka + 1];
    acc = wmma_f32_k4(av, bv, acc);
  }
#pragma unroll
  for (int v = 0; v < 8; ++v) {
    const int m = v + 8 * lh;
    E[m * D_ + e0 + l16] = acc[v] + b_d2[e0 + l16];
  }
  __syncthreads();

  if (tid < P_) {
    float s = b_attn[0];
    for (int e = 0; e < D_; ++e) s += E[tid * D_ + e] * w_attn[e];
    al[tid] = (mask[b * P_ + tid] == 0) ? -1e9f : s;
  }
  __syncthreads();
  if (tid < P_) {
    float mx = -1e30f;
#pragma unroll
    for (int q = 0; q < P_; ++q) mx = fmaxf(mx, al[q]);
    float sum = 0.f;
#pragma unroll
    for (int q = 0; q < P_; ++q) sum += __expf(al[q] - mx);
    aw[tid] = __expf(al[tid] - mx) / sum;
  }
  __syncthreads();
  if (tid < D_) {
    float s = 0.f;
#pragma unroll
    for (int q = 0; q < P_; ++q) s += aw[q] * E[q * D_ + tid];
    dctx[((size_t)bp * D_ + tid) * T_ + t] = s;   // stored as [B,P,D,T]
  }
}

// ===========================================================================
// Host-side orchestration
// ===========================================================================
extern "C" void kernel_launch(void* const* d_in, const int* in_sizes, int n_in,
                              void* d_out, int out_size, void* d_ws, size_t ws_size,
                              hipStream_t stream) {
  // ---- locate x / distances / padding_mask by (unique) element counts ----
  int ix = -1, idst = -1, imask = -1;
  for (int i = 0; i < n_in; ++i) {
    const int s = in_sizes[i];
    if (s == NTOT && ix < 0) ix = i;
    else if (s == B_ * T_ * P_ * P_ && idst < 0) idst = i;
    else if (s == B_ * P_ && imask < 0) imask = i;
  }
  int pbase = 0;
  for (int i = 0; i < n_in; ++i)
    if (i != ix && i != idst && i != imask) { pbase = i; break; }

  const float* x    = (const float*)d_in[ix];
  const float* dist = (const float*)d_in[idst];
  const int*   mask = (const int*)d_in[imask];
  auto F = [&](int i) { return (const float*)d_in[i]; };

  // params flattened (JAX sorted-key order): gcns(4x4), mlps(8x8), styls(4x20)
  // gcn j:  +0 Ws0, +1 Ws1, +2 bs0, +3 bs1
  // mlp i:  +0 a0, +1 a1, +2 b0, +3 b1, +4 b_s, +5 b_t, +6 w_s, +7 w_t
  // styl j: +0 a_n +1 a_ng +2 b_attn +3 b_d1 +4 b_d2 +5 b_emb +6 b_g +7 b_n
  //         +8 b_ng +9 b_out +10 b_pool +11 b_tmp +12 w_attn +13 w_d1 +14 w_d2
  //         +15 w_emb +16 w_g +17 w_out +18 w_pool +19 w_tmp
  const int GBASE = pbase;
  const int MBASE = pbase + 16;
  const int SBASE = pbase + 16 + 64;

  // ---- workspace layout (floats) ----
  float* ws    = (float*)d_ws;
  float* XBUF  = ws;
  float* XG    = ws + (size_t)NTOT;
  float* T1    = ws + (size_t)2 * NTOT;
  float* T2    = ws + (size_t)3 * NTOT;
  float* ADJ   = ws + (size_t)4 * NTOT;                  // 115200
  float* SCALE = ADJ + (size_t)B_ * T_ * P_ * P_;        // 204800
  float* SHIFT = SCALE + (size_t)B_ * D_ * T_;           // 204800
  float* ATTN  = SHIFT + (size_t)B_ * D_ * T_;           // 192

  hipMemcpyAsync(XBUF, x, (size_t)NTOT * sizeof(float),
                 hipMemcpyDeviceToDevice, stream);

  // adjacency softmax is layer-independent: compute once
  adj_softmax_kernel<<<(B_ * T_ * P_ + 255) / 256, 256, 0, stream>>>(dist, mask, ADJ);

  const int NBP = B_ * P_;   // 192 blocks

  for (int i = 0; i < 8; ++i) {
    const int mb = MBASE + i * 8;
    // temporal FC (WMMA) + LN, residual outside
    tfc_ln_kernel<<<NBP, 256, 0, stream>>>(
        XBUF, XBUF, F(mb + 7), F(mb + 5), F(mb + 0), F(mb + 2), XBUF, 0);
    // spatial FC (WMMA) + LN, residual outside
    spatial_fc_ln_kernel<<<NBP, 256, 0, stream>>>(
        XBUF, F(mb + 6), F(mb + 4), F(mb + 1), F(mb + 3), XBUF);

    if ((i + 1) % 2 == 0) {
      const int j  = (i + 1) / 2 - 1;
      const int gb = GBASE + j * 4;
      const int sb = SBASE + j * 20;

      // GCN: two WMMA layers (XBUF -> T1 -> XG), mask applied at the end
      gcn_layer_kernel<<<NBP, 256, 0, stream>>>(
          XBUF, ADJ, F(gb + 0), F(gb + 2), mask, T1, 0);
      gcn_layer_kernel<<<NBP, 256, 0, stream>>>(
          T1, ADJ, F(gb + 1), F(gb + 3), mask, XG, 1);

      // stylize: pooling -> scale/shift (must run on pre-update XG)
      attn_pool_kernel<<<B_, 256, 0, stream>>>(
          XG, F(sb + 18), F(sb + 10), mask, ATTN);
      scale_shift_kernel<<<B_, 256, 0, stream>>>(
          XG, ATTN, F(sb + 15), F(sb + 5), SCALE, SHIFT);

      // xg = LN(xg + x_clone @ w_g^T + b_g)   (XBUF still holds x_clone)
      tfc_ln_kernel<<<NBP, 256, 0, stream>>>(
          XG, XBUF, F(sb + 16), F(sb + 6), F(sb + 1), F(sb + 8), XG, 1);

      // x1 = (x*(1+scale)+shift) @ w_out^T + b_out  -> T1
      modulate_fc_kernel<<<NBP, 256, 0, stream>>>(
          XBUF, SCALE, SHIFT, F(sb + 17), F(sb + 9), T1);

      // fused distance-encoding GEMM + attention context -> T2 [B,P,D,T]
      denc_kernel<<<B_ * P_ * T_, 512, 0, stream>>>(
          dist, mask, F(sb + 13), F(sb + 3), F(sb + 14), F(sb + 4),
          F(sb + 12), F(sb + 2), T2);

      // x = LN(x1 + d_ctx @ w_tmp^T + b_tmp) * a_n + b_n  -> XBUF
      tfc_ln_kernel<<<NBP, 256, 0, stream>>>(
          T1, T2, F(sb + 19), F(sb + 11), F(sb + 0), F(sb + 7), XBUF, 1);
    }
  }

  hipMemcpyAsync(d_out, XBUF, (size_t)NTOT * sizeof(float),
                 hipMemcpyDeviceToDevice, stream);
  (void)out_size; (void)ws_size;
}